// MessagePassing_57097295233646
// MI455X (gfx1250) — compile-verified
//
#include <hip/hip_runtime.h>
#include <math.h>

// ---------------------------------------------------------------------------
// GraphSAGE forward for MI455X (gfx1250), FP32 end-to-end.
//   h = tanh(x @ W_in + b_in)
//   repeat L=3: agg = mean_{dst} h[src]; h = h@W_self + agg@W_neigh + b (+relu)
// GEMMs use V_WMMA_F32_16X16X4_F32 (wave32, 16x16 tile per wave).
// ---------------------------------------------------------------------------

typedef __attribute__((ext_vector_type(2))) float v2f;
typedef __attribute__((ext_vector_type(8))) float v8f;

#define N_NODES 200000
#define N_EDGES 600000
#define F_IN    117
#define F_PAD   124   // K-pad for LDS: multiple of 4; 124 mod 64 = -4 -> 16 rows hit 16 distinct banks
#define D       128
#define D_PAD   132   // 132 mod 64 = 4 -> conflict-free column reads across 16 rows
#define L_LAYERS 3

__device__ __forceinline__ v8f wmma_f32_16x16x4(v2f a, v2f b, v8f c) {
  // 8-arg form: (neg_a, A, neg_b, B, c_mod, C, reuse_a, reuse_b)
  return __builtin_amdgcn_wmma_f32_16x16x4_f32(false, a, false, b, (short)0, c,
                                               false, false);
}

// ---------------------------------------------------------------------------
// Degree / mean normalization
// ---------------------------------------------------------------------------
__global__ __launch_bounds__(256) void degree_kernel(const int* __restrict__ dst,
                                                     float* __restrict__ deg,
                                                     int nE) {
  int e = blockIdx.x * blockDim.x + threadIdx.x;
  if (e < nE) atomicAdd(&deg[dst[e]], 1.0f);
}

__global__ __launch_bounds__(256) void inv_deg_kernel(float* __restrict__ deg, int n) {
  int i = blockIdx.x * blockDim.x + threadIdx.x;
  if (i < n) deg[i] = 1.0f / fmaxf(deg[i], 1.0f);
}

// ---------------------------------------------------------------------------
// Input projection: h = tanh(x[N,117] @ W_in[117,128] + b_in)
// Workgroup = 16 rows x 128 cols; 8 waves, one 16x16 WMMA tile each.
// A tile staged in LDS (zero-padded K), B streamed from global (L2 resident).
// ---------------------------------------------------------------------------
__global__ __launch_bounds__(256) void input_gemm_tanh(const float* __restrict__ x,
                                                       const float* __restrict__ W_in,
                                                       const float* __restrict__ b_in,
                                                       float* __restrict__ h_out) {
  __shared__ float sx[16 * F_PAD];
  const int tid = threadIdx.x;
  const int mBase = blockIdx.x * 16;

  for (int i = tid; i < 16 * F_PAD; i += 256) {
    int r = i / F_PAD, c = i - r * F_PAD;
    sx[i] = (c < F_IN) ? x[(size_t)(mBase + r) * F_IN + c] : 0.0f;
  }
  __syncthreads();

  const int lane = tid & 31;
  const int half = lane >> 4;       // 0: lanes 0-15, 1: lanes 16-31
  const int l16  = lane & 15;
  const int col  = (tid >> 5) * 16 + l16;   // wave id selects 16-col tile

  v8f acc = {0.f, 0.f, 0.f, 0.f, 0.f, 0.f, 0.f, 0.f};
  for (int k = 0; k < F_IN; k += 4) {
    const int ka = k + 2 * half;    // A: lanes16-31 hold K=k+2,k+3 ; B mirrors
    v2f a, b;
    a.x = sx[l16 * F_PAD + ka];
    a.y = sx[l16 * F_PAD + ka + 1];
    b.x = (ka     < F_IN) ? W_in[(size_t)(ka)     * D + col] : 0.0f;
    b.y = (ka + 1 < F_IN) ? W_in[(size_t)(ka + 1) * D + col] : 0.0f;
    acc = wmma_f32_16x16x4(a, b, acc);
  }

  const float bias = b_in[col];
  const int rowBase = mBase + 8 * half;     // C: VGPR r -> M = r + 8*half
#pragma unroll
  for (int r = 0; r < 8; ++r)
    h_out[(size_t)(rowBase + r) * D + col] = tanhf(acc[r] + bias);
}

// ---------------------------------------------------------------------------
// Edge scatter: agg[dst] += h[src]   (one wave per edge, float4 per lane)
// ---------------------------------------------------------------------------
__global__ __launch_bounds__(256) void scatter_sum(const float* __restrict__ h,
                                                   const int* __restrict__ src,
                                                   const int* __restrict__ dst,
                                                   float* __restrict__ agg, int nE) {
  int e = blockIdx.x * 8 + (threadIdx.x >> 5);
  if (e >= nE) return;
  int lane = threadIdx.x & 31;
  int s = src[e], d = dst[e];
  const float4* hv = (const float4*)(h + (size_t)s * D);
  float4 v = hv[lane];                       // 32 lanes x 4 = 128 features
  float* ap = agg + (size_t)d * D + lane * 4;
  atomicAdd(ap + 0, v.x);
  atomicAdd(ap + 1, v.y);
  atomicAdd(ap + 2, v.z);
  atomicAdd(ap + 3, v.w);
}

// ---------------------------------------------------------------------------
// SAGE layer: out = h @ Wself + (agg * inv_deg) @ Wneigh + bias (+relu)
// Two WMMA accumulations per K-step into one 16x16 f32 tile.
// ---------------------------------------------------------------------------
__global__ __launch_bounds__(256) void layer_gemm(const float* __restrict__ h,
                                                  const float* __restrict__ agg,
                                                  const float* __restrict__ inv_deg,
                                                  const float* __restrict__ Wself,
                                                  const float* __restrict__ Wneigh,
                                                  const float* __restrict__ bias_v,
                                                  float* __restrict__ h_out,
                                                  int relu) {
  __shared__ float sh[16 * D_PAD];
  __shared__ float sa[16 * D_PAD];
  const int tid = threadIdx.x;
  const int mBase = blockIdx.x * 16;

  for (int i = tid; i < 16 * D; i += 256) {
    int r = i >> 7, c = i & (D - 1);
    int row = mBase + r;
    float id = inv_deg[row];
    sh[r * D_PAD + c] = h[(size_t)row * D + c];
    sa[r * D_PAD + c] = agg[(size_t)row * D + c] * id;   // fold mean here
  }
  __syncthreads();

  const int lane = tid & 31;
  const int half = lane >> 4;
  const int l16  = lane & 15;
  const int col  = (tid >> 5) * 16 + l16;

  v8f acc = {0.f, 0.f, 0.f, 0.f, 0.f, 0.f, 0.f, 0.f};
  for (int k = 0; k < D; k += 4) {
    const int ka = k + 2 * half;
    v2f a1, b1, a2, b2;
    a1.x = sh[l16 * D_PAD + ka];
    a1.y = sh[l16 * D_PAD + ka + 1];
    b1.x = Wself[(size_t)(ka)     * D + col];
    b1.y = Wself[(size_t)(ka + 1) * D + col];
    acc = wmma_f32_16x16x4(a1, b1, acc);

    a2.x = sa[l16 * D_PAD + ka];
    a2.y = sa[l16 * D_PAD + ka + 1];
    b2.x = Wneigh[(size_t)(ka)     * D + col];
    b2.y = Wneigh[(size_t)(ka + 1) * D + col];
    acc = wmma_f32_16x16x4(a2, b2, acc);
  }

  const float bv = bias_v[col];
  const int rowBase = mBase + 8 * half;
#pragma unroll
  for (int r = 0; r < 8; ++r) {
    float v = acc[r] + bv;
    if (relu) v = fmaxf(v, 0.0f);
    h_out[(size_t)(rowBase + r) * D + col] = v;
  }
}

// ---------------------------------------------------------------------------
// Orchestration (graph-capture safe: kernels + hipMemsetAsync only)
// Inputs: x, edge_index, W_in, b_in, W_self, W_neigh, b_layers
// ---------------------------------------------------------------------------
extern "C" void kernel_launch(void* const* d_in, const int* in_sizes, int n_in,
                              void* d_out, int out_size, void* d_ws, size_t ws_size,
                              hipStream_t stream) {
  const float* x       = (const float*)d_in[0];
  const int*   ei      = (const int*)d_in[1];     // [2, E]; row0=src, row1=dst
  const float* W_in    = (const float*)d_in[2];
  const float* b_in    = (const float*)d_in[3];
  const float* W_self  = (const float*)d_in[4];   // [L, D, D]
  const float* W_neigh = (const float*)d_in[5];   // [L, D, D]
  const float* b_lay   = (const float*)d_in[6];   // [L, D]
  float* out = (float*)d_out;

  float* ws   = (float*)d_ws;
  float* hA   = ws;                                   // [N, D]
  float* hB   = ws + (size_t)1 * N_NODES * D;         // [N, D]
  float* agg  = ws + (size_t)2 * N_NODES * D;         // [N, D]
  float* degv = ws + (size_t)3 * N_NODES * D;         // [N] deg -> inv_deg in place

  const int* src = ei;
  const int* dst = ei + N_EDGES;

  // degree + inverse (reused across layers)
  hipMemsetAsync(degv, 0, N_NODES * sizeof(float), stream);
  degree_kernel<<<(N_EDGES + 255) / 256, 256, 0, stream>>>(dst, degv, N_EDGES);
  inv_deg_kernel<<<(N_NODES + 255) / 256, 256, 0, stream>>>(degv, N_NODES);

  // input projection + tanh
  input_gemm_tanh<<<N_NODES / 16, 256, 0, stream>>>(x, W_in, b_in, hA);

  const float* hcur = hA;
  for (int i = 0; i < L_LAYERS; ++i) {
    hipMemsetAsync(agg, 0, (size_t)N_NODES * D * sizeof(float), stream);
    scatter_sum<<<(N_EDGES + 7) / 8, 256, 0, stream>>>(hcur, src, dst, agg, N_EDGES);
    float* hnext = (i == L_LAYERS - 1) ? out : (hcur == hA ? hB : hA);
    layer_gemm<<<N_NODES / 16, 256, 0, stream>>>(
        hcur, agg, degv,
        W_self  + (size_t)i * D * D,
        W_neigh + (size_t)i * D * D,
        b_lay   + (size_t)i * D,
        hnext, (i < L_LAYERS - 1) ? 1 : 0);
    hcur = hnext;
  }
}